// TemporalInteraction_12275016532563
// MI455X (gfx1250) — compile-verified
//
#include <hip/hip_runtime.h>

#ifndef __has_builtin
#define __has_builtin(x) 0
#endif

// Problem geometry (fixed by the reference): x is (B=8, C=128, T=8, H=56, W=56) f32.
// HW plane = 3136 floats = 784 float4s, contiguous in BOTH input and output.
#define T_DIM   8
#define C_DIM   128
#define PLANE4  784              // float4s per (b,c,t) HW plane
#define BC_STR4 (T_DIM * PLANE4) // 6272 float4s per (b,c)

typedef float v4f __attribute__((ext_vector_type(4)));
// Exact pointee type the async builtin expects: GCC-style int vector of 16 bytes.
typedef int   v4i __attribute__((vector_size(4 * sizeof(int))));
typedef __attribute__((address_space(1))) v4i g_v4i;   // global (AS1) int4
typedef __attribute__((address_space(3))) v4i l_v4i;   // LDS    (AS3) int4

#if __has_builtin(__builtin_amdgcn_global_load_async_to_lds_b128)
#define USE_ASYNC 1
#else
#define USE_ASYNC 0
#endif

__device__ __forceinline__ void wait_asynccnt0() {
#if __has_builtin(__builtin_amdgcn_s_wait_asynccnt)
    __builtin_amdgcn_s_wait_asynccnt(0);
#else
    asm volatile("s_wait_asynccnt 0" ::: "memory");
#endif
}

// ---------------------------------------------------------------------------
// Shared compute: out[t] = w0*v[t-1] + w1*v[t] + w2*v[t+1] + bias  (zero-padded)
// then scattered to the transposed plane ((b*T + t)*C + c).
// ---------------------------------------------------------------------------
__device__ __forceinline__ void compute_and_store(
    const v4f v[T_DIM], float w0, float w1, float w2, float bs,
    v4f* __restrict__ o4, unsigned b, unsigned c, unsigned col /*float4 idx in plane*/) {
#pragma unroll
    for (int t = 0; t < T_DIM; ++t) {
        v4f acc = w1 * v[t] + bs;
        if (t > 0)         acc += w0 * v[t - 1];
        if (t < T_DIM - 1) acc += w2 * v[t + 1];
        __builtin_nontemporal_store(
            acc, &o4[(((size_t)b * T_DIM + t) * C_DIM + c) * PLANE4 + col]);
    }
}

#if USE_ASYNC
// ---------------------------------------------------------------------------
// CDNA5 path: stage each 256-wide float4 chunk of all 8 T-planes into LDS with
// the async global->LDS data mover (tracked by ASYNCcnt), then compute from LDS.
// LDS footprint: 8 * 256 * 16B = 32 KB per workgroup (well under 320 KB/WGP).
// ---------------------------------------------------------------------------
__global__ __launch_bounds__(256) void temporal_shift_async(
    const float* __restrict__ x, const float* __restrict__ w,
    const float* __restrict__ bias, float* __restrict__ out) {
    __shared__ v4f tile[T_DIM][256];

    const unsigned tid = threadIdx.x;
    const unsigned bc  = blockIdx.x;           // 0 .. B*C-1 (1024)
    const unsigned b   = bc >> 7;              // /128
    const unsigned c   = bc & 127;
    const float w0 = w[0], w1 = w[1], w2 = w[2], bs = bias[0];

    const v4f* __restrict__ x4 = ((const v4f*)x) + (size_t)bc * BC_STR4;
    v4f* __restrict__ o4 = (v4f*)out;

    for (int s = 0; s < PLANE4; s += 256) {
        const int L = (PLANE4 - s) < 256 ? (PLANE4 - s) : 256;
        if (tid < (unsigned)L) {
#pragma unroll
            for (int t = 0; t < T_DIM; ++t) {
                const v4f* g = x4 + t * PLANE4 + s + tid;
                __builtin_amdgcn_global_load_async_to_lds_b128(
                    (g_v4i*)g, (l_v4i*)&tile[t][tid], /*offset=*/0, /*cpol=*/0);
            }
        }
        wait_asynccnt0();        // own wave's async copies landed in LDS
        __syncthreads();         // make all waves' chunks visible

        if (tid < (unsigned)L) {
            v4f v[T_DIM];
#pragma unroll
            for (int t = 0; t < T_DIM; ++t) v[t] = tile[t][tid];
            compute_and_store(v, w0, w1, w2, bs, o4, b, c, s + tid);
        }
        __syncthreads();         // LDS reuse fence before next chunk
    }
}
#endif // USE_ASYNC

// ---------------------------------------------------------------------------
// Fallback / baseline: direct b128 NT loads, one thread owns one float4 pixel
// column across all 8 T-planes (each input byte read exactly once).
// ---------------------------------------------------------------------------
__global__ __launch_bounds__(256) void temporal_shift_direct(
    const float* __restrict__ x, const float* __restrict__ w,
    const float* __restrict__ bias, float* __restrict__ out) {
    const unsigned gid = blockIdx.x * 256u + threadIdx.x;   // 0 .. 1024*784-1
    const unsigned bc  = gid / PLANE4;
    const unsigned col = gid - bc * PLANE4;
    const unsigned b   = bc >> 7;
    const unsigned c   = bc & 127;
    const float w0 = w[0], w1 = w[1], w2 = w[2], bs = bias[0];

    const v4f* __restrict__ x4 = ((const v4f*)x) + (size_t)bc * BC_STR4 + col;
    v4f* __restrict__ o4 = (v4f*)out;

    v4f v[T_DIM];
#pragma unroll
    for (int t = 0; t < T_DIM; ++t)
        v[t] = __builtin_nontemporal_load(&x4[t * PLANE4]);

    compute_and_store(v, w0, w1, w2, bs, o4, b, c, col);
}

// ---------------------------------------------------------------------------
extern "C" void kernel_launch(void* const* d_in, const int* in_sizes, int n_in,
                              void* d_out, int out_size, void* d_ws, size_t ws_size,
                              hipStream_t stream) {
    const float* x    = (const float*)d_in[0];   // 25,690,112 f32
    const float* w    = (const float*)d_in[1];   // 3 f32
    const float* bias = (const float*)d_in[2];   // 1 f32
    float* out = (float*)d_out;

    (void)in_sizes; (void)n_in; (void)out_size; (void)d_ws; (void)ws_size;

#if USE_ASYNC
    // One block per (b,c) pair: 1024 blocks x 8 wave32.
    temporal_shift_async<<<dim3(8 * C_DIM), dim3(256), 0, stream>>>(x, w, bias, out);
#else
    // 1024 * 784 float4 columns / 256 threads = 3136 blocks (exact).
    temporal_shift_direct<<<dim3((8 * C_DIM * PLANE4) / 256), dim3(256), 0, stream>>>(
        x, w, bias, out);
#endif
}